// GINLayer_6700148981977
// MI455X (gfx1250) — compile-verified
//
#include <hip/hip_runtime.h>

#define HID    128
#define NNODES 50000
#define NEDGES 640000
#define SROW   132            // LDS row stride (floats), padded: 132 % 64 = 4 banks
#define LN_EPS 1e-5f

typedef float v2f __attribute__((ext_vector_type(2)));
typedef float v8f __attribute__((ext_vector_type(8)));

// ---------------- kernel 1: z = (1+eps) * h ----------------
__global__ __launch_bounds__(256) void gin_init_z(const float* __restrict__ h,
                                                  const float* __restrict__ eps,
                                                  float* __restrict__ z) {
    const float s = 1.0f + eps[0];
    const float4* h4 = (const float4*)h;
    float4* z4 = (float4*)z;
    const int n4 = NNODES * HID / 4;
    for (int i = blockIdx.x * blockDim.x + threadIdx.x; i < n4;
         i += gridDim.x * blockDim.x) {
        float4 v = h4[i];
        v.x *= s; v.y *= s; v.z *= s; v.w *= s;
        z4[i] = v;
    }
}

// ---------------- kernel 2: z[dst] += h[src], one wave per edge ----------------
__global__ __launch_bounds__(256) void gin_scatter(const float* __restrict__ h,
                                                   const int* __restrict__ ei,
                                                   float* __restrict__ z) {
    int t = blockIdx.x * blockDim.x + threadIdx.x;
    int e = t >> 5;                 // wave32: one edge per wave
    if (e >= NEDGES) return;
    int lane = t & 31;              // 32 lanes x float4 = 128 floats
    int src = ei[e];
    int dst = ei[NEDGES + e];
    float4 v = ((const float4*)(h + (size_t)src * HID))[lane];
    float* zp = z + (size_t)dst * HID + lane * 4;
    atomicAdd(zp + 0, v.x);         // global_atomic_add_f32 (no return)
    atomicAdd(zp + 1, v.y);
    atomicAdd(zp + 2, v.z);
    atomicAdd(zp + 3, v.w);
}

// ---------------- kernel 3: fused MLP + residual + LayerNorm + ReLU ----------------
// block = 128 threads (4 waves), one 16-node tile per block.
__global__ __launch_bounds__(128) void gin_mlp_ln(
    const float* __restrict__ z,      // (1+eps)h + agg   [NNODES, HID]
    const float* __restrict__ h,      // residual input
    const float* __restrict__ W1, const float* __restrict__ b1,
    const float* __restrict__ W2, const float* __restrict__ b2,
    const float* __restrict__ gamma, const float* __restrict__ beta,
    float* __restrict__ out) {

    __shared__ float sA[16 * SROW];   // layer-1 input tile, reused for layer-2 output
    __shared__ float sZ1[16 * SROW];  // hidden activations

    const int t    = threadIdx.x;
    const int lane = t & 31;
    const int wave = t >> 5;
    const int node0 = blockIdx.x * 16;

    // ---- stage input tile (16 x 128) into LDS ----
    {
        int r  = t >> 3;             // 8 threads per row
        int c0 = (t & 7) * 16;
        const float* srcp = z + (size_t)(node0 + r) * HID + c0;
        float* dstp = sA + r * SROW + c0;
#pragma unroll
        for (int j = 0; j < 16; j += 4)
            *(float4*)(dstp + j) = *(const float4*)(srcp + j);
    }
    __syncthreads();

    const int m     = lane & 15;     // A: row m, lanes 16-31 carry k+2
    const int khalf = lane >> 4;
    const int n0a   = wave * 32;     // each wave owns two 16-col tiles
    const int n0b   = n0a + 16;

    // ---- layer 1: Z1 = relu(Zin @ W1^T + b1) ----
    v8f ca = {0.f,0.f,0.f,0.f,0.f,0.f,0.f,0.f};
    v8f cb = {0.f,0.f,0.f,0.f,0.f,0.f,0.f,0.f};
#pragma unroll
    for (int k0 = 0; k0 < HID; k0 += 4) {
        int kk = k0 + 2 * khalf;     // this lane's K pair
        v2f a  = *(const v2f*)(sA + m * SROW + kk);
        v2f wa = *(const v2f*)(W1 + (size_t)(n0a + m) * HID + kk); // B[k][n] = W1[n][k]
        v2f wb = *(const v2f*)(W1 + (size_t)(n0b + m) * HID + kk);
        ca = __builtin_amdgcn_wmma_f32_16x16x4_f32(false, a, false, wa, (short)0, ca, false, false);
        cb = __builtin_amdgcn_wmma_f32_16x16x4_f32(false, a, false, wb, (short)0, cb, false, false);
    }
    {
        float biasA = b1[n0a + m];
        float biasB = b1[n0b + m];
#pragma unroll
        for (int v = 0; v < 8; v++) {
            int row = v + 8 * khalf;                 // C/D layout
            float xa = ca[v] + biasA; xa = xa > 0.f ? xa : 0.f;
            float xb = cb[v] + biasB; xb = xb > 0.f ? xb : 0.f;
            sZ1[row * SROW + n0a + m] = xa;
            sZ1[row * SROW + n0b + m] = xb;
        }
    }
    __syncthreads();                 // all sA reads + sZ1 writes complete

    // ---- layer 2: Z2 = Z1 @ W2^T + b2  (into sA, now free) ----
    v8f da = {0.f,0.f,0.f,0.f,0.f,0.f,0.f,0.f};
    v8f db = {0.f,0.f,0.f,0.f,0.f,0.f,0.f,0.f};
#pragma unroll
    for (int k0 = 0; k0 < HID; k0 += 4) {
        int kk = k0 + 2 * khalf;
        v2f a  = *(const v2f*)(sZ1 + m * SROW + kk);
        v2f wa = *(const v2f*)(W2 + (size_t)(n0a + m) * HID + kk);
        v2f wb = *(const v2f*)(W2 + (size_t)(n0b + m) * HID + kk);
        da = __builtin_amdgcn_wmma_f32_16x16x4_f32(false, a, false, wa, (short)0, da, false, false);
        db = __builtin_amdgcn_wmma_f32_16x16x4_f32(false, a, false, wb, (short)0, db, false, false);
    }
    {
        float biasA = b2[n0a + m];
        float biasB = b2[n0b + m];
#pragma unroll
        for (int v = 0; v < 8; v++) {
            int row = v + 8 * khalf;
            sA[row * SROW + n0a + m] = da[v] + biasA;
            sA[row * SROW + n0b + m] = db[v] + biasB;
        }
    }
    __syncthreads();

    // ---- residual + LayerNorm + ReLU ----
    {
        int r  = t >> 3;             // 8 lanes per row -> width-8 shuffle reduce
        int c0 = (t & 7) * 16;
        const float* hrow = h + (size_t)(node0 + r) * HID + c0;
        float vals[16];
        float s = 0.f, s2 = 0.f;
#pragma unroll
        for (int j = 0; j < 16; j++) {
            float x = sA[r * SROW + c0 + j] + hrow[j];
            vals[j] = x;
            s += x; s2 += x * x;
        }
#pragma unroll
        for (int off = 1; off < 8; off <<= 1) {
            s  += __shfl_xor(s,  off, 8);
            s2 += __shfl_xor(s2, off, 8);
        }
        float mu  = s * (1.0f / HID);
        float var = s2 * (1.0f / HID) - mu * mu;
        float inv = rsqrtf(var + LN_EPS);
        float* orow = out + (size_t)(node0 + r) * HID + c0;
#pragma unroll
        for (int j = 0; j < 16; j++) {
            float y = (vals[j] - mu) * inv * gamma[c0 + j] + beta[c0 + j];
            orow[j] = y > 0.f ? y : 0.f;
        }
    }
}

extern "C" void kernel_launch(void* const* d_in, const int* in_sizes, int n_in,
                              void* d_out, int out_size, void* d_ws, size_t ws_size,
                              hipStream_t stream) {
    const float* h     = (const float*)d_in[0];
    const int*   ei    = (const int*)  d_in[1];
    const float* W1    = (const float*)d_in[2];
    const float* b1    = (const float*)d_in[3];
    const float* W2    = (const float*)d_in[4];
    const float* b2    = (const float*)d_in[5];
    const float* eps   = (const float*)d_in[6];
    const float* gamma = (const float*)d_in[7];
    const float* beta  = (const float*)d_in[8];
    float* out = (float*)d_out;
    float* z   = (float*)d_ws;        // NNODES*HID floats = 25.6 MB scratch

    gin_init_z<<<6250, 256, 0, stream>>>(h, eps, z);
    gin_scatter<<<(NEDGES * 32) / 256, 256, 0, stream>>>(h, ei, z);
    gin_mlp_ln<<<NNODES / 16, 128, 0, stream>>>(z, h, W1, b1, W2, b2, gamma, beta, out);
}